// ST_AffineDiff_34394098106556
// MI455X (gfx1250) — compile-verified
//
#include <hip/hip_runtime.h>

#define BATCH 32
#define CHN   3
#define HH    512
#define WW    512

typedef float v2f __attribute__((ext_vector_type(2)));
typedef float v8f __attribute__((ext_vector_type(8)));

// ---------------------------------------------------------------------------
// Kernel 1: per-batch 3x3 matrix exponential (scaling & squaring + Taylor-16)
// folded with affine_grid + grid_sample normalization into pixel-space affine:
//   ix = c0*w + c1*h + c2 ;  iy = c3*w + c4*h + c5
// One wave, one batch per lane. Writes 6 floats per batch into workspace.
// ---------------------------------------------------------------------------
__device__ __forceinline__ void mm3(const float a[3][3], const float b[3][3],
                                    float o[3][3]) {
#pragma unroll
  for (int i = 0; i < 3; ++i)
#pragma unroll
    for (int j = 0; j < 3; ++j)
      o[i][j] = a[i][0] * b[0][j] + a[i][1] * b[1][j] + a[i][2] * b[2][j];
}

__global__ void st_expm_coeff_kernel(const float* __restrict__ theta,
                                     float* __restrict__ cs) {
  const int b = threadIdx.x;
  if (b >= BATCH) return;

  float A[3][3];
#pragma unroll
  for (int j = 0; j < 3; ++j) {
    A[0][j] = theta[b * 6 + j];
    A[1][j] = theta[b * 6 + 3 + j];
    A[2][j] = 0.0f;
  }

  // inf-norm (last row is zero)
  const float n0 = fabsf(A[0][0]) + fabsf(A[0][1]) + fabsf(A[0][2]);
  const float n1 = fabsf(A[1][0]) + fabsf(A[1][1]) + fabsf(A[1][2]);
  float nrm = fmaxf(n0, n1);
  int s = 0;
  while (nrm > 0.5f && s < 30) { nrm *= 0.5f; ++s; }
  const float sc = ldexpf(1.0f, -s);

  float M[3][3], P[3][3], E[3][3], T[3][3];
#pragma unroll
  for (int i = 0; i < 3; ++i)
#pragma unroll
    for (int j = 0; j < 3; ++j) {
      M[i][j] = A[i][j] * sc;
      P[i][j] = (i == j) ? 1.0f : 0.0f;
      E[i][j] = (i == j) ? 1.0f : 0.0f;
    }

  // Taylor: E = sum_{k=0..16} M^k / k!   (||M|| <= 0.5 -> fp32-exact)
  for (int k = 1; k <= 16; ++k) {
    mm3(P, M, T);
    const float inv = 1.0f / (float)k;
#pragma unroll
    for (int i = 0; i < 3; ++i)
#pragma unroll
      for (int j = 0; j < 3; ++j) {
        P[i][j] = T[i][j] * inv;
        E[i][j] += P[i][j];
      }
  }
  // undo scaling: square s times
  for (int q = 0; q < s; ++q) {
    mm3(E, E, T);
#pragma unroll
    for (int i = 0; i < 3; ++i)
#pragma unroll
      for (int j = 0; j < 3; ++j) E[i][j] = T[i][j];
  }

  const float t00 = E[0][0], t01 = E[0][1], t02 = E[0][2];
  const float t10 = E[1][0], t11 = E[1][1], t12 = E[1][2];

  // xs = w*(2/W) + bw, ys = h*(2/H) + bh ; ix = (gx+1)*W/2 - 0.5 (align_corners=False)
  const float bw = 1.0f / (float)WW - 1.0f;
  const float bh = 1.0f / (float)HH - 1.0f;
  float* o = cs + b * 6;
  o[0] = t00;                                                    // c00 * w
  o[1] = t01 * ((float)WW / (float)HH);                          // c01 * h
  o[2] = (t00 * bw + t01 * bh + t02 + 1.0f) * (0.5f * WW) - 0.5f;
  o[3] = t10 * ((float)HH / (float)WW);                          // c10 * w
  o[4] = t11;                                                    // c11 * h
  o[5] = (t10 * bw + t11 * bh + t12 + 1.0f) * (0.5f * HH) - 0.5f;
}

// ---------------------------------------------------------------------------
// Kernel 2: one wave per 16x16 output tile. The affine-grid einsum is done on
// the matrix unit with V_WMMA_F32_16X16X4_F32 (K=2 used, K=2..3 zero):
//   D[m][n] = A[m][0]*B[0][n] + A[m][1]*B[1][n]
//   A[m][*] = { m, 1 } ; B[0][n] = c_h ; B[1][n] = c_w*n + base
// giving ix/iy for all 256 pixels of the tile in the C/D accumulator layout.
// Each lane then bilinear-samples its 8 pixels x 3 channels (gather via L2;
// whole 100MB input fits in the 192MB L2), NT-stores the output.
// ---------------------------------------------------------------------------
__global__ __launch_bounds__(256) void st_sample_kernel(
    const float* __restrict__ x, const float* __restrict__ cs,
    float* __restrict__ out) {
  const int lane = threadIdx.x & 31;
  const int wave = threadIdx.x >> 5;
  const int tile = blockIdx.x * 8 + wave;   // 32768 tiles total
  const int b    = tile >> 10;              // / (32*32 tiles per image)
  const int rem  = tile & 1023;
  const int h0   = (rem >> 5) << 4;
  const int w0   = (rem & 31) << 4;

  const float* c = cs + b * 6;
  const float c00 = c[0], c01 = c[1], c02 = c[2];
  const float c10 = c[3], c11 = c[4], c12 = c[5];
  const float basex = c00 * (float)w0 + c01 * (float)h0 + c02;
  const float basey = c10 * (float)w0 + c11 * (float)h0 + c12;

  // A (16x4 f32): lanes 0-15 hold row m=lane, VGPR0=K0, VGPR1=K1;
  //               lanes 16-31 hold K2/K3 (zeroed).
  // B (4x16 f32): VGPR0 = row K0 (lanes 0-15) / K2 (lanes 16-31),
  //               VGPR1 = row K1 / K3. K2,K3 rows are zero.
  const bool  lo = lane < 16;
  const float fl = (float)(lane & 15);
  v2f a, bx, by;
  a.x  = lo ? fl   : 0.0f;                  // A[m][0] = m
  a.y  = lo ? 1.0f : 0.0f;                  // A[m][1] = 1
  bx.x = lo ? c01  : 0.0f;                  // B[0][n] = c01
  bx.y = lo ? (c00 * fl + basex) : 0.0f;    // B[1][n] = c00*n + basex
  by.x = lo ? c11  : 0.0f;
  by.y = lo ? (c10 * fl + basey) : 0.0f;

  v8f cz = {};
  v8f ix8 = __builtin_amdgcn_wmma_f32_16x16x4_f32(false, a, false, bx,
                                                  (short)0, cz, false, false);
  v8f iy8 = __builtin_amdgcn_wmma_f32_16x16x4_f32(false, a, false, by,
                                                  (short)0, cz, false, false);

  // C/D layout: lane(0-15): VGPR v -> (row v, col lane); lane(16-31): (row v+8, col lane-16)
  const int n  = lane & 15;
  const int mb = lo ? 0 : 8;
  const size_t plane = (size_t)HH * WW;
  const float* img = x + (size_t)b * CHN * plane;
  float*       op  = out + (size_t)b * CHN * plane;

#pragma unroll
  for (int v = 0; v < 8; ++v) {
    const int hh = h0 + mb + v;
    const int wp = w0 + n;
    const float ix = ix8[v];
    const float iy = iy8[v];

    const float xf = floorf(ix);
    const float yf = floorf(iy);
    const float fx = ix - xf, fy = iy - yf;
    const float gx = 1.0f - fx, gy = 1.0f - fy;

    const bool vx0 = (xf >= 0.0f)  && (xf <= (float)(WW - 1));
    const bool vx1 = (xf >= -1.0f) && (xf <= (float)(WW - 2));
    const bool vy0 = (yf >= 0.0f)  && (yf <= (float)(HH - 1));
    const bool vy1 = (yf >= -1.0f) && (yf <= (float)(HH - 2));

    const float w00 = gx * gy * (float)(vx0 && vy0);   // (x0,y0)
    const float w01 = gx * fy * (float)(vx0 && vy1);   // (x0,y1)
    const float w10 = fx * gy * (float)(vx1 && vy0);   // (x1,y0)
    const float w11 = fx * fy * (float)(vx1 && vy1);   // (x1,y1)

    const int xi0 = (int)fminf(fmaxf(xf,        0.0f), (float)(WW - 1));
    const int yi0 = (int)fminf(fmaxf(yf,        0.0f), (float)(HH - 1));
    const int xi1 = (int)fminf(fmaxf(xf + 1.0f, 0.0f), (float)(WW - 1));
    const int yi1 = (int)fminf(fmaxf(yf + 1.0f, 0.0f), (float)(HH - 1));

    const int o00 = yi0 * WW + xi0, o01 = yi1 * WW + xi0;
    const int o10 = yi0 * WW + xi1, o11 = yi1 * WW + xi1;
    const size_t ob = (size_t)hh * WW + wp;

#pragma unroll
    for (int ch = 0; ch < CHN; ++ch) {
      const float* pl = img + ch * plane;
      const float r = w00 * pl[o00] + w01 * pl[o01] +
                      w10 * pl[o10] + w11 * pl[o11];
      __builtin_nontemporal_store(r, op + ch * plane + ob);  // keep L2 for input
    }
  }
}

extern "C" void kernel_launch(void* const* d_in, const int* in_sizes, int n_in,
                              void* d_out, int out_size, void* d_ws, size_t ws_size,
                              hipStream_t stream) {
  const float* x     = (const float*)d_in[0];   // [32,3,512,512] fp32
  const float* theta = (const float*)d_in[1];   // [32,6] fp32
  float* out = (float*)d_out;                   // [32,3,512,512] fp32
  float* cs  = (float*)d_ws;                    // 32*6 floats of scratch

  st_expm_coeff_kernel<<<1, 32, 0, stream>>>(theta, cs);

  const int tiles = BATCH * (HH / 16) * (WW / 16);      // 32768
  st_sample_kernel<<<tiles / 8, 256, 0, stream>>>(x, cs, out);
}